// EdgeFusionGCN_64072322122517
// MI455X (gfx1250) — compile-verified
//
#include <hip/hip_runtime.h>

typedef __attribute__((ext_vector_type(2))) float v2f;
typedef __attribute__((ext_vector_type(8))) float v8f;

#define N_NODES 100000
#define N_EDGES 1600000
#define D       128
#define N_REL   5
#define N_EFEAT 4

// ---------------------------------------------------------------------------
// Kernel 1: per-node transforms.
// grid.x = N/16 node tiles, grid.y = 11 weight matrices:
//   0..4  : W_rel[r]         -> xW[n][r][:]
//   5     : loop_rel (+b_rel)  -> acc[n][0:128]
//   6..9  : W_emb[r-6]       -> xWe[n][r-6][:]
//   10    : loop_emb (+b_emb) -> acc[n][128:256]
// 256 threads = 8 waves; wave w computes output columns [16w, 16w+16).
// Destination resolved to (base, per-node stride, bias) ONCE -> branch-free
// store loop.
// ---------------------------------------------------------------------------
__global__ __launch_bounds__(256) void transform_kernel(
    const float* __restrict__ nfeats,
    const float* __restrict__ W_rel, const float* __restrict__ loop_rel,
    const float* __restrict__ b_rel,
    const float* __restrict__ W_emb, const float* __restrict__ loop_emb,
    const float* __restrict__ b_emb,
    float* __restrict__ xW, float* __restrict__ xWe, float* __restrict__ acc)
{
    const int LSTR = 132;                 // 128 + 4 pad: bank-conflict-free rows
    __shared__ float As[16 * LSTR];

    const int tid      = threadIdx.x;
    const int nodeBase = blockIdx.x * 16;
    const int wid      = blockIdx.y;      // which weight matrix (uniform)

    // Stage 16x128 A tile (512 float4, 2 per thread), coalesced.
    for (int i = tid; i < 512; i += 256) {
        int row = i >> 5;                 // 32 float4 per row
        int c4  = i & 31;
        float4 val = *(const float4*)(nfeats + (size_t)(nodeBase + row) * D + c4 * 4);
        float* dst = &As[row * LSTR + c4 * 4];
        dst[0] = val.x; dst[1] = val.y; dst[2] = val.z; dst[3] = val.w;
    }
    __syncthreads();

    const int wave  = tid >> 5;
    const int lane  = tid & 31;
    const int mrow  = lane & 15;          // A row this lane feeds
    const int khalf = (lane >> 4) << 1;   // 0 (lanes 0-15) or 2 (lanes 16-31)
    const int col   = wave * 16 + (lane & 15);

    // Uniform selection of B matrix and output placement (resolved up front).
    const float* B;
    float*       dstBase;                 // already offset by col
    size_t       dstStride;               // floats per node
    float        bias = 0.0f;
    if (wid < 5) {
        B         = W_rel + (size_t)wid * D * D;
        dstBase   = xW + (size_t)wid * D + col;
        dstStride = (size_t)N_REL * D;
    } else if (wid == 5) {
        B         = loop_rel;
        dstBase   = acc + col;
        dstStride = (size_t)2 * D;
        bias      = b_rel[col];
    } else if (wid < 10) {
        B         = W_emb + (size_t)(wid - 6) * D * D;
        dstBase   = xWe + (size_t)(wid - 6) * D + col;
        dstStride = (size_t)N_EFEAT * D;
    } else {
        B         = loop_emb;
        dstBase   = acc + D + col;
        dstStride = (size_t)2 * D;
        bias      = b_emb[col];
    }

    v8f c = {};
    #pragma unroll 8
    for (int k = 0; k < D; k += 4) {
        const int kk = k + khalf;
        v2f a, b;
        a[0] = As[mrow * LSTR + kk];
        a[1] = As[mrow * LSTR + kk + 1];
        b[0] = B[(size_t)kk * D + col];
        b[1] = B[(size_t)(kk + 1) * D + col];
        c = __builtin_amdgcn_wmma_f32_16x16x4_f32(false, a, false, b,
                                                  (short)0, c, false, false);
    }

    // C/D layout: VGPR r -> M=r (lanes 0-15) or M=8+r (lanes 16-31); N = lane&15.
    const int mBase = (lane >> 4) * 8;
    #pragma unroll
    for (int r = 0; r < 8; ++r) {
        const size_t node = (size_t)(nodeBase + mBase + r);
        dstBase[node * dstStride] = c[r] + bias;
    }
}

// ---------------------------------------------------------------------------
// Kernel 2: edge gather/scatter. One wave per edge; lane owns 4 floats.
// v_etype += m * xW[u, etype]
// v_emb   += sum_r (efeats[e,r] + [r==emb_etype]) * xWe[u,r]
// mask==0 edges contribute nothing -> whole wave exits.
// ---------------------------------------------------------------------------
__global__ __launch_bounds__(256) void edge_kernel(
    const float* __restrict__ efeats,
    const int* __restrict__ uarr, const int* __restrict__ varr,
    const int* __restrict__ etypes, const int* __restrict__ embt,
    const int* __restrict__ mask,
    const float* __restrict__ xW, const float* __restrict__ xWe,
    float* __restrict__ acc)
{
    // Wave-uniform edge index -> scalar loads for the per-edge scalars.
    const int e = __builtin_amdgcn_readfirstlane(
        (int)(blockIdx.x * 8 + (threadIdx.x >> 5)));
    if (e >= N_EDGES) return;
    if (mask[e] == 0) return;

    const int u  = uarr[e];
    const int v  = varr[e];
    const int et = etypes[e];
    const int r0 = embt[e];

    float c0 = efeats[(size_t)e * 4 + 0];
    float c1 = efeats[(size_t)e * 4 + 1];
    float c2 = efeats[(size_t)e * 4 + 2];
    float c3 = efeats[(size_t)e * 4 + 3];
    if      (r0 == 0) c0 += 1.0f;
    else if (r0 == 1) c1 += 1.0f;
    else if (r0 == 2) c2 += 1.0f;
    else              c3 += 1.0f;

    const int lane = threadIdx.x & 31;

    const float4* p1 = (const float4*)(xW + ((size_t)u * N_REL + et) * D);
    float4 v1 = p1[lane];

    const float4* pe = (const float4*)(xWe + (size_t)u * N_EFEAT * D);
    float4 a0 = pe[lane];
    float4 a1 = pe[32 + lane];
    float4 a2 = pe[64 + lane];
    float4 a3 = pe[96 + lane];

    float4 v2;
    v2.x = c0 * a0.x + c1 * a1.x + c2 * a2.x + c3 * a3.x;
    v2.y = c0 * a0.y + c1 * a1.y + c2 * a2.y + c3 * a3.y;
    v2.z = c0 * a0.z + c1 * a1.z + c2 * a2.z + c3 * a3.z;
    v2.w = c0 * a0.w + c1 * a1.w + c2 * a2.w + c3 * a3.w;

    float* d = acc + (size_t)v * 2 * D + lane * 4;
    atomicAdd(d + 0, v1.x);
    atomicAdd(d + 1, v1.y);
    atomicAdd(d + 2, v1.z);
    atomicAdd(d + 3, v1.w);
    atomicAdd(d + D + 0, v2.x);
    atomicAdd(d + D + 1, v2.y);
    atomicAdd(d + D + 2, v2.z);
    atomicAdd(d + D + 3, v2.w);
}

// ---------------------------------------------------------------------------
// Kernel 3: out[n] = acc[n, 0:256] @ fc_w[256,128]. Same WMMA tiling, K=256.
// ---------------------------------------------------------------------------
__global__ __launch_bounds__(256) void out_kernel(
    const float* __restrict__ acc, const float* __restrict__ fc_w,
    float* __restrict__ out)
{
    const int K    = 2 * D;               // 256
    const int LSTR = 260;                 // 256 + 4 pad
    __shared__ float As[16 * LSTR];

    const int tid      = threadIdx.x;
    const int nodeBase = blockIdx.x * 16;

    for (int i = tid; i < 1024; i += 256) {   // 16 rows x 64 float4
        int row = i >> 6;
        int c4  = i & 63;
        float4 val = *(const float4*)(acc + (size_t)(nodeBase + row) * K + c4 * 4);
        float* dst = &As[row * LSTR + c4 * 4];
        dst[0] = val.x; dst[1] = val.y; dst[2] = val.z; dst[3] = val.w;
    }
    __syncthreads();

    const int wave  = tid >> 5;
    const int lane  = tid & 31;
    const int mrow  = lane & 15;
    const int khalf = (lane >> 4) << 1;
    const int col   = wave * 16 + (lane & 15);

    v8f c = {};
    #pragma unroll 8
    for (int k = 0; k < K; k += 4) {
        const int kk = k + khalf;
        v2f a, b;
        a[0] = As[mrow * LSTR + kk];
        a[1] = As[mrow * LSTR + kk + 1];
        b[0] = fc_w[(size_t)kk * D + col];
        b[1] = fc_w[(size_t)(kk + 1) * D + col];
        c = __builtin_amdgcn_wmma_f32_16x16x4_f32(false, a, false, b,
                                                  (short)0, c, false, false);
    }

    const int mBase = (lane >> 4) * 8;
    #pragma unroll
    for (int r = 0; r < 8; ++r) {
        out[(size_t)(nodeBase + mBase + r) * D + col] = c[r];
    }
}

// ---------------------------------------------------------------------------
extern "C" void kernel_launch(void* const* d_in, const int* in_sizes, int n_in,
                              void* d_out, int out_size, void* d_ws, size_t ws_size,
                              hipStream_t stream)
{
    const float* nfeats   = (const float*)d_in[0];
    const float* efeats   = (const float*)d_in[1];
    const int*   u        = (const int*)  d_in[2];
    const int*   v        = (const int*)  d_in[3];
    const int*   etypes   = (const int*)  d_in[4];
    const int*   embt     = (const int*)  d_in[5];
    const int*   mask     = (const int*)  d_in[6];
    const float* W_rel    = (const float*)d_in[7];
    const float* loop_rel = (const float*)d_in[8];
    const float* b_rel    = (const float*)d_in[9];
    const float* W_emb    = (const float*)d_in[10];
    const float* loop_emb = (const float*)d_in[11];
    const float* b_emb    = (const float*)d_in[12];
    const float* fc_w     = (const float*)d_in[13];
    float*       out      = (float*)d_out;

    // Workspace layout (floats): acc[N][256] | xW[N][5][128] | xWe[N][4][128]
    float* acc = (float*)d_ws;
    float* xW  = acc + (size_t)N_NODES * 2 * D;
    float* xWe = xW  + (size_t)N_NODES * N_REL * D;

    dim3 g1(N_NODES / 16, 11);
    transform_kernel<<<g1, 256, 0, stream>>>(nfeats, W_rel, loop_rel, b_rel,
                                             W_emb, loop_emb, b_emb,
                                             xW, xWe, acc);

    edge_kernel<<<N_EDGES / 8, 256, 0, stream>>>(efeats, u, v, etypes, embt,
                                                 mask, xW, xWe, acc);

    out_kernel<<<N_NODES / 16, 256, 0, stream>>>(acc, fc_w, out);
}